// QuantumStateEncoder_13941463842875
// MI455X (gfx1250) — compile-verified
//
#include <hip/hip_runtime.h>
#include <hip/hip_bf16.h>
#include <math.h>

typedef float v2f __attribute__((ext_vector_type(2)));
typedef float v8f __attribute__((ext_vector_type(8)));

#define N_QUBITS 8
#define N_LAYERS 2
#define DIM      256           // 2^N_QUBITS
#define NB       16            // samples per workgroup (WMMA N tile)
#define PSI_STRIDE 260         // padded LDS row stride (floats): 260 % 64 == 4 -> conflict free

// ---------------------------------------------------------------------------
// Kernel A: build the 256x256 complex unitary of the StronglyEntanglingLayers.
// One wave (32 lanes) per basis column; state = 256 complex amps, 8 per lane.
// amp index bits: [7:3] = lane, [2:0] = local register t. Qubit q <-> bit 7-q.
// ---------------------------------------------------------------------------

template <int P>
__device__ __forceinline__ void apply1q(float (&ar)[8], float (&ai)[8], int lane,
                                        float u00r, float u00i, float u01r, float u01i,
                                        float u10r, float u10i, float u11r, float u11i) {
  if constexpr (P < 3) {
    constexpr int st = 1 << P;
#pragma unroll
    for (int t = 0; t < 8; ++t) {
      if (!(t & st)) {
        float a0r = ar[t], a0i = ai[t];
        float a1r = ar[t | st], a1i = ai[t | st];
        ar[t]      = u00r * a0r - u00i * a0i + u01r * a1r - u01i * a1i;
        ai[t]      = u00r * a0i + u00i * a0r + u01r * a1i + u01i * a1r;
        ar[t | st] = u10r * a0r - u10i * a0i + u11r * a1r - u11i * a1i;
        ai[t | st] = u10r * a0i + u10i * a0r + u11r * a1i + u11i * a1r;
      }
    }
  } else {
    constexpr int mask = 1 << (P - 3);
    const bool hi = (lane & mask) != 0;
    // hi lane: new = u10*partner + u11*mine ; lo lane: new = u00*mine + u01*partner
    const float c0r = hi ? u10r : u00r, c0i = hi ? u10i : u00i;
    const float c1r = hi ? u11r : u01r, c1i = hi ? u11i : u01i;
#pragma unroll
    for (int t = 0; t < 8; ++t) {
      float pr = __shfl_xor(ar[t], mask, 32);
      float pi = __shfl_xor(ai[t], mask, 32);
      float a0r = hi ? pr : ar[t], a0i = hi ? pi : ai[t];
      float a1r = hi ? ar[t] : pr, a1i = hi ? ai[t] : pi;
      ar[t] = c0r * a0r - c0i * a0i + c1r * a1r - c1i * a1i;
      ai[t] = c0r * a0i + c0i * a0r + c1r * a1i + c1i * a1r;
    }
  }
}

__device__ __forceinline__ void rot_gate(float (&ar)[8], float (&ai)[8], int lane, int p,
                                         float u00r, float u00i, float u01r, float u01i,
                                         float u10r, float u10i, float u11r, float u11i) {
  switch (p) {
    case 0: apply1q<0>(ar, ai, lane, u00r, u00i, u01r, u01i, u10r, u10i, u11r, u11i); break;
    case 1: apply1q<1>(ar, ai, lane, u00r, u00i, u01r, u01i, u10r, u10i, u11r, u11i); break;
    case 2: apply1q<2>(ar, ai, lane, u00r, u00i, u01r, u01i, u10r, u10i, u11r, u11i); break;
    case 3: apply1q<3>(ar, ai, lane, u00r, u00i, u01r, u01i, u10r, u10i, u11r, u11i); break;
    case 4: apply1q<4>(ar, ai, lane, u00r, u00i, u01r, u01i, u10r, u10i, u11r, u11i); break;
    case 5: apply1q<5>(ar, ai, lane, u00r, u00i, u01r, u01i, u10r, u10i, u11r, u11i); break;
    case 6: apply1q<6>(ar, ai, lane, u00r, u00i, u01r, u01i, u10r, u10i, u11r, u11i); break;
    default: apply1q<7>(ar, ai, lane, u00r, u00i, u01r, u01i, u10r, u10i, u11r, u11i); break;
  }
}

template <int PT>
__device__ __forceinline__ void cnotT(float (&ar)[8], float (&ai)[8], int lane, int pc) {
  if constexpr (PT < 3) {
    constexpr int st = 1 << PT;
#pragma unroll
    for (int t = 0; t < 8; ++t) {
      if (!(t & st)) {
        int idx = (lane << 3) | t;       // ctrl bit identical for t and t|st (pc != PT)
        bool ctrl = ((idx >> pc) & 1) != 0;
        float t0r = ar[t], t0i = ai[t];
        float t1r = ar[t | st], t1i = ai[t | st];
        ar[t]      = ctrl ? t1r : t0r;
        ai[t]      = ctrl ? t1i : t0i;
        ar[t | st] = ctrl ? t0r : t1r;
        ai[t | st] = ctrl ? t0i : t1i;
      }
    }
  } else {
    constexpr int mask = 1 << (PT - 3);
#pragma unroll
    for (int t = 0; t < 8; ++t) {
      float pr = __shfl_xor(ar[t], mask, 32);
      float pi = __shfl_xor(ai[t], mask, 32);
      int idx = (lane << 3) | t;
      bool ctrl = ((idx >> pc) & 1) != 0;
      ar[t] = ctrl ? pr : ar[t];
      ai[t] = ctrl ? pi : ai[t];
    }
  }
}

__device__ __forceinline__ void cnot_gate(float (&ar)[8], float (&ai)[8], int lane, int pc, int pt) {
  switch (pt) {
    case 0: cnotT<0>(ar, ai, lane, pc); break;
    case 1: cnotT<1>(ar, ai, lane, pc); break;
    case 2: cnotT<2>(ar, ai, lane, pc); break;
    case 3: cnotT<3>(ar, ai, lane, pc); break;
    case 4: cnotT<4>(ar, ai, lane, pc); break;
    case 5: cnotT<5>(ar, ai, lane, pc); break;
    case 6: cnotT<6>(ar, ai, lane, pc); break;
    default: cnotT<7>(ar, ai, lane, pc); break;
  }
}

__global__ __launch_bounds__(32) void build_unitary_kernel(const float* __restrict__ qw,
                                                           float* __restrict__ Ur,
                                                           float* __restrict__ Ui) {
  const int lane = threadIdx.x;   // 0..31
  const int col  = blockIdx.x;    // basis column 0..255
  float ar[8], ai[8];
#pragma unroll
  for (int t = 0; t < 8; ++t) { ar[t] = 0.f; ai[t] = 0.f; }
  if ((col >> 3) == lane) ar[col & 7] = 1.0f;

#pragma unroll
  for (int l = 0; l < N_LAYERS; ++l) {
#pragma unroll
    for (int q = 0; q < N_QUBITS; ++q) {
      const float* g = qw + (l * N_QUBITS + q) * 3;
      float phi = g[0], th = g[1], om = g[2];
      float ch = cosf(0.5f * th), sh = sinf(0.5f * th);
      float hp = 0.5f * (phi + om), hm = 0.5f * (phi - om);
      float cp = cosf(hp), sp = sinf(hp);
      float cm = cosf(hm), sm = sinf(hm);
      // Rot = [[e^{-i hp} c, -e^{+i hm} s], [e^{-i hm} s, e^{+i hp} c]]
      rot_gate(ar, ai, lane, 7 - q,
               cp * ch, -sp * ch,    // u00
               -cm * sh, -sm * sh,   // u01
               cm * sh, -sm * sh,    // u10
               cp * ch, sp * ch);    // u11
    }
    int r = (l % (N_QUBITS - 1)) + 1;
#pragma unroll
    for (int q = 0; q < N_QUBITS; ++q) {
      cnot_gate(ar, ai, lane, 7 - q, 7 - ((q + r) & 7));
    }
  }
#pragma unroll
  for (int t = 0; t < 8; ++t) {
    int m = (lane << 3) | t;
    Ur[m * DIM + col] = ar[t];
    Ui[m * DIM + col] = ai[t];
  }
}

// ---------------------------------------------------------------------------
// Kernel B: fused forward. 16 samples per block, 256 threads (8 waves).
// psi product-state -> phi = U*psi via V_WMMA_F32_16X16X4_F32 -> <Z_q> -> MLP.
// ---------------------------------------------------------------------------
__global__ __launch_bounds__(256) void fused_forward_kernel(
    const float* __restrict__ x, const float* __restrict__ Wp, const float* __restrict__ bp,
    const float* __restrict__ W1, const float* __restrict__ b1,
    const float* __restrict__ W2, const float* __restrict__ b2,
    const float* __restrict__ Ur, const float* __restrict__ Ui,
    float* __restrict__ out) {
  __shared__ float cs_c[NB][N_QUBITS];
  __shared__ float cs_s[NB][N_QUBITS];
  __shared__ float z_lds[NB][N_QUBITS];
  __shared__ float psi_rt[NB][PSI_STRIDE];   // transposed: [n][k]
  __shared__ float psi_it[NB][PSI_STRIDE];
  __shared__ float h_lds[NB][64];

  const int tid = threadIdx.x;
  const int sbase = blockIdx.x * NB;

  // --- 1. angles = x @ Wp^T + bp ; precompute cos/sin(angle/2) -------------
  if (tid < NB * N_QUBITS) {
    int n = tid >> 3, q = tid & 7;
    const float* xr = x + (size_t)(sbase + n) * 17;
    const float* wr = Wp + q * 17;
    float a = bp[q];
#pragma unroll
    for (int d = 0; d < 17; ++d) a += xr[d] * wr[d];
    a *= 0.5f;
    cs_c[n][q] = cosf(a);
    cs_s[n][q] = sinf(a);
    z_lds[n][q] = 0.f;
  }
  __syncthreads();

  // --- 2. product state psi (RX embedding of |0..0>) -----------------------
  // amp(idx) = prod_q (bit? sin : cos) * (-i)^popcount(idx)
#pragma unroll 1
  for (int it = 0; it < NB; ++it) {
    int task = it * 256 + tid;
    int n = task >> 8;
    int idx = task & 255;
    float p = 1.f;
#pragma unroll
    for (int q = 0; q < N_QUBITS; ++q)
      p *= ((idx >> (7 - q)) & 1) ? cs_s[n][q] : cs_c[n][q];
    int k = __popc(idx) & 3;
    float pr = (k == 0) ? p : ((k == 2) ? -p : 0.f);
    float pi = (k == 1) ? -p : ((k == 3) ? p : 0.f);
    psi_rt[n][idx] = pr;
    psi_it[n][idx] = pi;
  }
  __syncthreads();

  // --- 3. phi = U * psi (complex f32 GEMM via WMMA), probs, <Z_q> ----------
  {
    const int wave = tid >> 5;
    const int lane = tid & 31;
    const int nloc = lane & 15;      // N index (sample within tile)
    const int half = lane >> 4;      // K half select per ISA A/B layout
    const int m0a = (wave * 2 + 0) * 16;
    const int m0b = (wave * 2 + 1) * 16;

    const float* ur_a = Ur + (size_t)(m0a + nloc) * DIM + half * 2;
    const float* ui_a = Ui + (size_t)(m0a + nloc) * DIM + half * 2;
    const float* ur_b = Ur + (size_t)(m0b + nloc) * DIM + half * 2;
    const float* ui_b = Ui + (size_t)(m0b + nloc) * DIM + half * 2;
    const float* ppr  = &psi_rt[nloc][half * 2];
    const float* ppi  = &psi_it[nloc][half * 2];

    v8f arr0 = {}, aii0 = {}, aim0 = {};   // tile a: Ur*Pr, Ui*Pi, (Ur*Pi + Ui*Pr)
    v8f arr1 = {}, aii1 = {}, aim1 = {};   // tile b

#pragma unroll 2
    for (int kb = 0; kb < DIM; kb += 4) {
      v2f bpr  = *(const v2f*)(ppr + kb);
      v2f bpi  = *(const v2f*)(ppi + kb);
      v2f aur0 = *(const v2f*)(ur_a + kb);
      v2f aui0 = *(const v2f*)(ui_a + kb);
      v2f aur1 = *(const v2f*)(ur_b + kb);
      v2f aui1 = *(const v2f*)(ui_b + kb);
      arr0 = __builtin_amdgcn_wmma_f32_16x16x4_f32(false, aur0, false, bpr, (short)0, arr0, false, false);
      aii0 = __builtin_amdgcn_wmma_f32_16x16x4_f32(false, aui0, false, bpi, (short)0, aii0, false, false);
      aim0 = __builtin_amdgcn_wmma_f32_16x16x4_f32(false, aur0, false, bpi, (short)0, aim0, false, false);
      aim0 = __builtin_amdgcn_wmma_f32_16x16x4_f32(false, aui0, false, bpr, (short)0, aim0, false, false);
      arr1 = __builtin_amdgcn_wmma_f32_16x16x4_f32(false, aur1, false, bpr, (short)0, arr1, false, false);
      aii1 = __builtin_amdgcn_wmma_f32_16x16x4_f32(false, aui1, false, bpi, (short)0, aii1, false, false);
      aim1 = __builtin_amdgcn_wmma_f32_16x16x4_f32(false, aur1, false, bpi, (short)0, aim1, false, false);
      aim1 = __builtin_amdgcn_wmma_f32_16x16x4_f32(false, aui1, false, bpr, (short)0, aim1, false, false);
    }

    // C/D layout: vgpr r -> M = m0 + half*8 + r, lane -> N = lane&15
    float zp[N_QUBITS];
#pragma unroll
    for (int q = 0; q < N_QUBITS; ++q) zp[q] = 0.f;
#pragma unroll
    for (int r = 0; r < 8; ++r) {
      {
        float fr = arr0[r] - aii0[r], fi = aim0[r];
        float prob = fr * fr + fi * fi;
        int m = m0a + half * 8 + r;
#pragma unroll
        for (int q = 0; q < N_QUBITS; ++q)
          zp[q] += ((m >> (7 - q)) & 1) ? -prob : prob;
      }
      {
        float fr = arr1[r] - aii1[r], fi = aim1[r];
        float prob = fr * fr + fi * fi;
        int m = m0b + half * 8 + r;
#pragma unroll
        for (int q = 0; q < N_QUBITS; ++q)
          zp[q] += ((m >> (7 - q)) & 1) ? -prob : prob;
      }
    }
#pragma unroll
    for (int q = 0; q < N_QUBITS; ++q)
      atomicAdd(&z_lds[nloc][q], zp[q]);   // ds_add_f32
  }
  __syncthreads();

  // --- 4. h = relu(z @ W1^T + b1) ------------------------------------------
  for (int i = tid; i < NB * 64; i += 256) {
    int n = i >> 6, j = i & 63;
    float a = b1[j];
#pragma unroll
    for (int q = 0; q < N_QUBITS; ++q) a += W1[j * N_QUBITS + q] * z_lds[n][q];
    h_lds[n][j] = fmaxf(a, 0.f);
  }
  __syncthreads();

  // --- 5. out = tanh(h @ W2^T + b2) ----------------------------------------
  for (int i = tid; i < NB * 32; i += 256) {
    int n = i >> 5, o = i & 31;
    float a = b2[o];
#pragma unroll
    for (int j = 0; j < 64; ++j) a += W2[o * 64 + j] * h_lds[n][j];
    out[(size_t)(sbase + n) * 32 + o] = tanhf(a);
  }
}

// ---------------------------------------------------------------------------
extern "C" void kernel_launch(void* const* d_in, const int* in_sizes, int n_in,
                              void* d_out, int out_size, void* d_ws, size_t ws_size,
                              hipStream_t stream) {
  const float* x  = (const float*)d_in[0];
  const float* Wp = (const float*)d_in[1];
  const float* bp = (const float*)d_in[2];
  const float* qw = (const float*)d_in[3];
  const float* W1 = (const float*)d_in[4];
  const float* b1 = (const float*)d_in[5];
  const float* W2 = (const float*)d_in[6];
  const float* b2 = (const float*)d_in[7];

  float* Ur = (float*)d_ws;                 // 256*256 floats
  float* Ui = Ur + DIM * DIM;               // 256*256 floats (total 512 KB)

  build_unitary_kernel<<<DIM, 32, 0, stream>>>(qw, Ur, Ui);

  const int B = in_sizes[0] / 17;           // 65536
  fused_forward_kernel<<<B / NB, 256, 0, stream>>>(x, Wp, bp, W1, b1, W2, b2, Ur, Ui,
                                                   (float*)d_out);
}